// HybridAttention_55559696941626
// MI455X (gfx1250) — compile-verified
//
#include <hip/hip_runtime.h>
#include <math.h>

// ---------------- problem constants ----------------
#define BATCH 8
#define CH    128
#define HH    256
#define WW    256
#define HWSZ  (HH*WW)            // 65536
#define NPIX  (BATCH*HWSZ)       // 524288
#define NELEM ((size_t)BATCH*CH*HWSZ) // 67108864
#define OUTLL 135                // dwt LL output length per axis
#define PADLEN 285               // padded-then-dropped length per axis

typedef __attribute__((ext_vector_type(2))) float v2f;
typedef __attribute__((ext_vector_type(8))) float v8f;

__constant__ float DB8_DEC_LO[16] = {
    -0.00011747678400228192f, 0.0006754494059985568f, -0.0003917403729959771f,
    -0.00487035299301066f, 0.008746094047015655f, 0.013981027917015516f,
    -0.04408825393106472f, -0.01736930100202211f, 0.128747426620186f,
    0.00047248457399797254f, -0.2840155429624281f, -0.015829105256023893f,
    0.5853546836548691f, 0.6756307362980128f, 0.3128715909144659f,
    0.05441584224308161f
};

__device__ __forceinline__ float sigmoidf_(float x) {
    return 1.0f / (1.0f + __expf(-x));
}

// ---------------------------------------------------------------------------
// Kernel 1: build folded dwt weight vector u[256].
// For axis length 256: symmetric pad 15 both sides, drop first sample, then
// stride-2 valid correlation with f = reverse(dec_lo), 135 outputs.
// g[j] = total filter weight landing on padded sample j; fold symmetric
// padding back to original index n -> u[n].
// ---------------------------------------------------------------------------
__global__ void k_uweights(float* __restrict__ u) {
    int n = threadIdx.x;   // 0..255
    float acc = 0.0f;
    for (int j = 0; j < PADLEN; ++j) {
        int i = j + 1;  // index in the (pre-drop) 286-length padded array
        int orig = (i < 15) ? (14 - i) : ((i <= 270) ? (i - 15) : (526 - i));
        if (orig != n) continue;
        float g = 0.0f;
        #pragma unroll
        for (int k = 0; k < 16; ++k) {
            int d = j - k;                       // = 2*o
            if (d >= 0 && (d & 1) == 0 && (d >> 1) <= (OUTLL - 1))
                g += DB8_DEC_LO[15 - k];         // f[k] = dec_lo reversed
        }
        acc += g;
    }
    u[n] = acc;
}

// ---------------------------------------------------------------------------
// Kernel 2: per-pixel mean & max over the 128 channels.
// ---------------------------------------------------------------------------
__global__ void k_avgmax(const float* __restrict__ x,
                         float* __restrict__ avg, float* __restrict__ mx) {
    int p  = blockIdx.x * blockDim.x + threadIdx.x;  // 0..NPIX-1
    int b  = p >> 16;
    int hw = p & (HWSZ - 1);
    const float* xp = x + (size_t)b * CH * HWSZ + hw;
    float s = 0.0f, m = -3.402823466e38f;
    #pragma unroll 4
    for (int c = 0; c < CH; ++c) {
        if (c + 2 < CH) __builtin_prefetch(xp + (size_t)(c + 2) * HWSZ, 0, 0);
        float v = xp[(size_t)c * HWSZ];
        s += v;
        m = fmaxf(m, v);
    }
    avg[p] = s * (1.0f / CH);
    mx[p]  = m;
}

// ---------------------------------------------------------------------------
// Kernel 3: y[b,c] = (1/135^2) * sum_{h,w} x[b,c,h,w] * u[h] * u[w]
// One block per (b,c); thread t owns column w = t.
// ---------------------------------------------------------------------------
__global__ void k_chansum(const float* __restrict__ x,
                          const float* __restrict__ u,
                          float* __restrict__ y) {
    __shared__ float ush[HH];
    __shared__ float red[256];
    int t = threadIdx.x;
    ush[t] = u[t];
    __syncthreads();
    int bc = blockIdx.x;                         // 0..1023
    const float* xp = x + (size_t)bc * HWSZ;
    float acc = 0.0f;
    for (int h = 0; h < HH; ++h)
        acc += xp[h * WW + t] * ush[h];
    acc *= ush[t];                               // column weight u[w]
    red[t] = acc;
    __syncthreads();
    for (int s = 128; s > 0; s >>= 1) {
        if (t < s) red[t] += red[t + s];
        __syncthreads();
    }
    if (t == 0) y[bc] = red[0] * (1.0f / (135.0f * 135.0f));
}

// ---------------------------------------------------------------------------
// Kernel 4: 7x7 conv over {avg, max} (zero pad 3) + sigmoid -> attn.
// conv_w layout [1,2,7,7]: first 49 taps for avg, next 49 for max.
// ---------------------------------------------------------------------------
__global__ void k_conv(const float* __restrict__ avg, const float* __restrict__ mx,
                       const float* __restrict__ cw, float* __restrict__ attn) {
    __shared__ float wsh[98];
    if (threadIdx.x < 98) wsh[threadIdx.x] = cw[threadIdx.x];
    __syncthreads();
    int p = blockIdx.x * blockDim.x + threadIdx.x;
    int b = p >> 16;
    int h = (p >> 8) & 255;
    int w = p & 255;
    const float* sa = avg + (size_t)b * HWSZ;
    const float* sm = mx  + (size_t)b * HWSZ;
    float acc = 0.0f;
    #pragma unroll
    for (int kh = 0; kh < 7; ++kh) {
        int hh = h + kh - 3;
        if (hh < 0 || hh >= HH) continue;
        #pragma unroll
        for (int kw = 0; kw < 7; ++kw) {
            int wc = w + kw - 3;
            if (wc < 0 || wc >= WW) continue;
            int o = hh * WW + wc;
            acc = fmaf(sa[o], wsh[kh * 7 + kw], acc);
            acc = fmaf(sm[o], wsh[49 + kh * 7 + kw], acc);
        }
    }
    attn[p] = sigmoidf_(acc);
}

// ---------------------------------------------------------------------------
// Kernel 5: FC layers via V_WMMA_F32_16X16X4_F32 (single wave).
//   h = relu(y @ fc_w1.T)          [8x8]   (M=8 pad 16, K=128, N=8 pad 16)
//   w = sigmoid(h @ fc_w2.T)       [8x128] (M=8 pad 16, K=8,  N=128 in 8 tiles)
// Operands are staged zero-padded in LDS so every WMMA input load is an
// unconditional ds_load_b64 (no exec-mask predication in the hot loop).
// f32 A 16x4 layout: lanes 0-15 -> K={0,1}, lanes 16-31 -> K={2,3} (2 VGPRs).
// B 4x16 analogous (row striped across lanes).  D: lane = N, vgpr r -> M=r+8*half.
// ---------------------------------------------------------------------------
__global__ void k_fc(const float* __restrict__ y,  const float* __restrict__ fw1,
                     const float* __restrict__ fw2, const float* __restrict__ wt,
                     float* __restrict__ wout, float* __restrict__ alpha_out) {
    __shared__ __align__(16) float ysh[16 * CH];   // A1: rows 8..15 zero
    __shared__ __align__(16) float w1sh[16 * CH];  // B1: cols 8..15 zero
    __shared__ __align__(16) float hsh[16 * 8];    // A2: rows 8..15 zero
    int lane = threadIdx.x;                        // 0..31
    int mn   = lane & 15;                          // row of A / col of B
    int khalf = lane >> 4;                         // 0 or 1

    // stage zero-padded operands (1024 real + 1024 zero each)
    for (int i = lane; i < 16 * CH; i += 32) {
        float yv = (i < 8 * CH) ? y[i]   : 0.0f;
        float wv = (i < 8 * CH) ? fw1[i] : 0.0f;
        ysh[i]  = yv;
        w1sh[i] = wv;
    }
    for (int i = lane; i < 16 * 8; i += 32) hsh[i] = 0.0f;
    __syncthreads();

    // ---- FC1: [16x128] x [128x16] via 32 k-steps of 16x16x4 ----
    v8f c1 = {};
    for (int kk = 0; kk < 32; ++kk) {
        int k0 = kk * 4 + khalf * 2;               // even -> 8B aligned
        v2f a = *(const v2f*)&ysh[mn * CH + k0];   // A[m][k]  = y (padded)
        v2f b = *(const v2f*)&w1sh[mn * CH + k0];  // B[k][n] = fc_w1[n][k]
        c1 = __builtin_amdgcn_wmma_f32_16x16x4_f32(false, a, false, b,
                                                   (short)0, c1, false, false);
    }
    if (khalf == 0 && mn < 8) {
        #pragma unroll
        for (int r = 0; r < 8; ++r)                // rows 0..7 live in half 0
            hsh[r * 8 + mn] = fmaxf(c1[r], 0.0f);
    }
    __syncthreads();

    // ---- FC2: [16x8] x [8x128] -> 8 N-tiles, 2 k-steps each ----
    for (int t = 0; t < 8; ++t) {
        v8f c2 = {};
        #pragma unroll
        for (int ks = 0; ks < 2; ++ks) {
            int k0 = ks * 4 + khalf * 2;           // 0..6, always < K=8
            v2f a = *(const v2f*)&hsh[mn * 8 + k0];
            int n = t * 16 + mn;                   // 0..127, all valid
            v2f b = *(const v2f*)&fw2[n * 8 + k0]; // B[k][n] = fc_w2[n][k]
            c2 = __builtin_amdgcn_wmma_f32_16x16x4_f32(false, a, false, b,
                                                       (short)0, c2, false, false);
        }
        if (khalf == 0) {
            #pragma unroll
            for (int r = 0; r < 8; ++r)            // batch rows 0..7
                wout[r * CH + t * 16 + mn] = sigmoidf_(c2[r]);
        }
    }
    if (lane == 0) alpha_out[0] = sigmoidf_(wt[0]);
}

// ---------------------------------------------------------------------------
// Kernel 6: out = x * (alpha*attn + (1-alpha)*w), float4 vectorized.
// ---------------------------------------------------------------------------
__global__ void k_final(const float* __restrict__ x, const float* __restrict__ attn,
                        const float* __restrict__ wcoef, const float* __restrict__ wt,
                        float* __restrict__ out) {
    size_t i4 = (size_t)blockIdx.x * blockDim.x + threadIdx.x;
    size_t e  = i4 * 4;                          // element offset, multiple of 4
    int bc = (int)(e >> 16);                     // b*128 + c
    int b  = bc >> 7;
    int hw = (int)(e & (HWSZ - 1));
    float alpha = sigmoidf_(wt[0]);
    float wv = wcoef[bc];
    const float4 x4 = *(const float4*)(x + e);
    const float4 a4 = *(const float4*)(attn + (size_t)b * HWSZ + hw);
    float4 o;
    o.x = x4.x * (alpha * a4.x + (1.0f - alpha) * wv);
    o.y = x4.y * (alpha * a4.y + (1.0f - alpha) * wv);
    o.z = x4.z * (alpha * a4.z + (1.0f - alpha) * wv);
    o.w = x4.w * (alpha * a4.w + (1.0f - alpha) * wv);
    *(float4*)(out + e) = o;
}

// ---------------------------------------------------------------------------
extern "C" void kernel_launch(void* const* d_in, const int* in_sizes, int n_in,
                              void* d_out, int out_size, void* d_ws, size_t ws_size,
                              hipStream_t stream) {
    const float* x   = (const float*)d_in[0];   // [8,128,256,256]
    const float* cw  = (const float*)d_in[1];   // [1,2,7,7]
    const float* fw1 = (const float*)d_in[2];   // [8,128]
    const float* fw2 = (const float*)d_in[3];   // [128,8]
    const float* wt  = (const float*)d_in[4];   // scalar

    // d_out layout: out | attn | w | alpha
    float* out_p   = (float*)d_out;
    float* attn_p  = out_p + NELEM;                      // +67108864
    float* w_p     = attn_p + (size_t)BATCH * HWSZ;      // +524288
    float* alpha_p = w_p + (size_t)BATCH * CH;           // +1024

    // workspace layout
    float* ws    = (float*)d_ws;
    float* avg_p = ws;                                   // 524288
    float* mx_p  = ws + (size_t)NPIX;                    // 524288
    float* u_p   = ws + (size_t)2 * NPIX;                // 256
    float* y_p   = u_p + 256;                            // 1024

    k_uweights<<<1, 256, 0, stream>>>(u_p);
    k_avgmax  <<<NPIX / 256, 256, 0, stream>>>(x, avg_p, mx_p);
    k_chansum <<<BATCH * CH, 256, 0, stream>>>(x, u_p, y_p);
    k_conv    <<<NPIX / 256, 256, 0, stream>>>(avg_p, mx_p, cw, attn_p);
    k_fc      <<<1, 32, 0, stream>>>(y_p, fw1, fw2, wt, w_p, alpha_p);
    k_final   <<<(unsigned)(NELEM / 4 / 256), 256, 0, stream>>>(x, attn_p, w_p, wt, out_p);
}